// CriticGNN_16784732192967
// MI455X (gfx1250) — compile-verified
//
#include <hip/hip_runtime.h>
#include <hip/hip_bf16.h>
#include <math.h>

// ---------------------------------------------------------------------------
// CriticGNN forward on gfx1250 (MI455X, wave32, WMMA).
//   3x GATConv(+self-loops, segment-softmax) -> BN+ReLU -> mean-pool -> MLP.
// Dense matmuls use V_WMMA_F32_16X16X4_F32 (fp32 matrix core path; GEMM is
// not the bottleneck so we keep full precision). Edge scatter work is the
// bandwidth-bound part but the 51 MB feature matrix fits in the 192 MB L2,
// so gathers/atomics stay on-chip.
// ---------------------------------------------------------------------------

#define Nn 100000
#define Ee 1600000
#define ETOT (Ee + Nn)
#define FEAT 64
#define HID 128

typedef float v2f __attribute__((ext_vector_type(2)));
typedef float v8f __attribute__((ext_vector_type(8)));

// ---- monotone float <-> uint key (for atomic segment-max without CAS) -----
__device__ __forceinline__ unsigned f2key(float f) {
    unsigned u = __float_as_uint(f);
    return (u & 0x80000000u) ? ~u : (u | 0x80000000u);
}
__device__ __forceinline__ float key2f(unsigned k) {
    unsigned u = (k & 0x80000000u) ? (k & 0x7fffffffu) : ~k;
    return __uint_as_float(u);
}

// ---------------------------------------------------------------------------
// fills
// ---------------------------------------------------------------------------
__global__ void fill_f(float* __restrict__ p, float v, size_t n) {
    size_t i = (size_t)blockIdx.x * blockDim.x + threadIdx.x;
    size_t st = (size_t)gridDim.x * blockDim.x;
    for (; i < n; i += st) p[i] = v;
}
__global__ void fill_u(unsigned* __restrict__ p, unsigned v, size_t n) {
    size_t i = (size_t)blockIdx.x * blockDim.x + threadIdx.x;
    size_t st = (size_t)gridDim.x * blockDim.x;
    for (; i < n; i += st) p[i] = v;
}

// ---------------------------------------------------------------------------
// H = X @ W   (X: [nrows x K] row-major, W: [K x 128] row-major, H: [nrows x 128])
// One block = 256 threads = 8 waves; block covers 16 rows, wave w covers
// columns [16w, 16w+16). K stepped by 4 with V_WMMA_F32_16X16X4_F32.
//   A VGPR layout (ISA 7.12.2): lane L holds A[M=L&15][K = khalf + i], i=0,1
//     where khalf = 2*(L>>4).
//   B assumed symmetric: VGPR i holds B[K=khalf+i][N=L&15].
//   D: acc[v] = D[M = v + 8*(L>>4)][N = L&15].
// ---------------------------------------------------------------------------
__global__ void gemm_wmma_f32(const float* __restrict__ X,
                              const float* __restrict__ W,
                              float* __restrict__ H, int nrows, int K) {
    const int wave = threadIdx.x >> 5;
    const int lane = threadIdx.x & 31;
    const int r0 = blockIdx.x * 16;
    if (r0 >= nrows) return;                 // block-uniform: EXEC stays all-1s
    const int c0 = wave * 16;
    const int mrow = lane & 15;
    const int khalf = (lane >> 4) << 1;      // 0 or 2
    const float* xrow = X + (size_t)(r0 + mrow) * K + khalf;

    v8f acc = {};
    for (int k0 = 0; k0 < K; k0 += 4) {
        v2f a, b;
        a.x = xrow[k0];
        a.y = xrow[k0 + 1];
        b.x = W[(size_t)(k0 + khalf) * HID + c0 + mrow];
        b.y = W[(size_t)(k0 + khalf + 1) * HID + c0 + mrow];
        acc = __builtin_amdgcn_wmma_f32_16x16x4_f32(
            /*neg_a=*/false, a, /*neg_b=*/false, b,
            /*c_mod=*/(short)0, acc, /*reuse_a=*/false, /*reuse_b=*/false);
    }
    const int hi = lane >> 4;
#pragma unroll
    for (int v = 0; v < 8; ++v) {
        int m = v + 8 * hi;
        H[(size_t)(r0 + m) * HID + c0 + (lane & 15)] = acc[v];
    }
}

// ---------------------------------------------------------------------------
// per-node attention logits: als[n,h] = <H[n, hC:(h+1)C], a_src[h]>, same ald.
// one wave per node.
// ---------------------------------------------------------------------------
__global__ void att_dots(const float* __restrict__ Hf,
                         const float* __restrict__ a_s,
                         const float* __restrict__ a_d,
                         float* __restrict__ als, float* __restrict__ ald,
                         int n, int heads) {
    int wid = (int)((blockIdx.x * (size_t)blockDim.x + threadIdx.x) >> 5);
    int lane = threadIdx.x & 31;
    if (wid >= n) return;
    const int C = HID / heads;
    const float* row = Hf + (size_t)wid * HID;
    for (int h = 0; h < heads; ++h) {
        float s = 0.f, d = 0.f;
        for (int c = lane; c < C; c += 32) {
            float x = row[h * C + c];
            s += x * a_s[h * C + c];
            d += x * a_d[h * C + c];
        }
        for (int off = 16; off; off >>= 1) {
            s += __shfl_xor(s, off);
            d += __shfl_xor(d, off);
        }
        if (lane == 0) {
            als[wid * heads + h] = s;
            ald[wid * heads + h] = d;
        }
    }
}

// ---------------------------------------------------------------------------
// edge pass 1: segment-max of leaky_relu(als[src]+ald[dst]) into mkey[dst,h]
// ---------------------------------------------------------------------------
__global__ void edge_logit_max(const int* __restrict__ ei,
                               const float* __restrict__ als,
                               const float* __restrict__ ald,
                               unsigned* __restrict__ mkey, int hshift) {
    const int Hh = 1 << hshift;
    const long long total = (long long)ETOT << hshift;
    long long i = (long long)blockIdx.x * blockDim.x + threadIdx.x;
    const long long st = (long long)gridDim.x * blockDim.x;
    for (; i < total; i += st) {
        int e = (int)(i >> hshift);
        int h = (int)(i & (Hh - 1));
        int s = (e < Ee) ? ei[e] : (e - Ee);
        int d = (e < Ee) ? ei[Ee + e] : (e - Ee);
        float v = als[s * Hh + h] + ald[d * Hh + h];
        v = (v > 0.f) ? v : 0.2f * v;                 // LeakyReLU(0.2)
        atomicMax(&mkey[d * Hh + h], f2key(v));
    }
}

// ---------------------------------------------------------------------------
// edge pass 2: ex = exp(e - m[dst]); den[dst] += ex; alpha[e,h] = ex
// ---------------------------------------------------------------------------
__global__ void edge_exp_den(const int* __restrict__ ei,
                             const float* __restrict__ als,
                             const float* __restrict__ ald,
                             const unsigned* __restrict__ mkey,
                             float* __restrict__ den,
                             float* __restrict__ alpha, int hshift) {
    const int Hh = 1 << hshift;
    const long long total = (long long)ETOT << hshift;
    long long i = (long long)blockIdx.x * blockDim.x + threadIdx.x;
    const long long st = (long long)gridDim.x * blockDim.x;
    for (; i < total; i += st) {
        int e = (int)(i >> hshift);
        int h = (int)(i & (Hh - 1));
        int s = (e < Ee) ? ei[e] : (e - Ee);
        int d = (e < Ee) ? ei[Ee + e] : (e - Ee);
        float v = als[s * Hh + h] + ald[d * Hh + h];
        v = (v > 0.f) ? v : 0.2f * v;
        float m = key2f(mkey[d * Hh + h]);
        if (!(m > -3.0e38f && m < 3.0e38f)) m = 0.f;  // mirror where(isfinite(m),m,0)
        float ex = __expf(v - m);
        atomicAdd(&den[d * Hh + h], ex);
        alpha[i] = ex;                                 // i == e*Hh + h
    }
}

// ---------------------------------------------------------------------------
// edge pass 3: out[dst, :] += (alpha/den[dst]) * H[src, :]   (thread per (e,c))
// ---------------------------------------------------------------------------
__global__ void edge_aggregate(const int* __restrict__ ei,
                               const float* __restrict__ Hf,
                               const float* __restrict__ alpha,
                               const float* __restrict__ den,
                               float* __restrict__ out, int cshift) {
    const int Hh = HID >> cshift;
    const long long total = (long long)ETOT << 7;      // ETOT * 128
    long long i = (long long)blockIdx.x * blockDim.x + threadIdx.x;
    const long long st = (long long)gridDim.x * blockDim.x;
    for (; i < total; i += st) {
        int e = (int)(i >> 7);
        int c = (int)(i & 127);
        int h = c >> cshift;
        int s = (e < Ee) ? ei[e] : (e - Ee);
        int d = (e < Ee) ? ei[Ee + e] : (e - Ee);
        float a = alpha[(size_t)e * Hh + h] / (den[d * Hh + h] + 1e-16f);
        atomicAdd(&out[(size_t)d * HID + c], a * Hf[(size_t)s * HID + c]);
    }
}

// ---------------------------------------------------------------------------
// BatchNorm: per-channel partial sums, finalize to scale/shift, apply+ReLU
// ---------------------------------------------------------------------------
__global__ void col_stats(const float* __restrict__ X, float* __restrict__ accum,
                          int n, int do_sq) {
    int c = threadIdx.x;                              // 128 threads
    float s = 0.f, q = 0.f;
    for (int r = blockIdx.x; r < n; r += gridDim.x) {
        float v = X[(size_t)r * HID + c];
        s += v;
        q += v * v;
    }
    atomicAdd(&accum[c], s);
    if (do_sq) atomicAdd(&accum[HID + c], q);
}

__global__ void bn_finalize(const float* __restrict__ accum,
                            const float* __restrict__ g,
                            const float* __restrict__ b,
                            float* __restrict__ sc, float* __restrict__ sh,
                            float inv_n) {
    int c = threadIdx.x;
    float mean = accum[c] * inv_n;
    float var = accum[HID + c] * inv_n - mean * mean;  // biased var, matches x.var(0)
    float s = g[c] * rsqrtf(var + 1e-5f);
    sc[c] = s;
    sh[c] = b[c] - mean * s;
}

__global__ void bn_apply_relu(float* __restrict__ X,
                              const float* __restrict__ sc,
                              const float* __restrict__ sh, int n) {
    size_t total = (size_t)n * HID;
    size_t i = (size_t)blockIdx.x * blockDim.x + threadIdx.x;
    size_t st = (size_t)gridDim.x * blockDim.x;
    for (; i < total; i += st) {
        int c = (int)(i & 127);
        X[i] = fmaxf(sc[c] * X[i] + sh[c], 0.f);
    }
}

// ---------------------------------------------------------------------------
// value head: g = pool/N; v = relu(g@vW1+vb1)@vW2 + vb2   (single block)
// ---------------------------------------------------------------------------
__global__ void value_head(const float* __restrict__ pool,
                           const float* __restrict__ vW1,
                           const float* __restrict__ vb1,
                           const float* __restrict__ vW2,
                           const float* __restrict__ vb2,
                           float* __restrict__ out, float inv_n) {
    __shared__ float g[HID];
    __shared__ float red[HID];
    int j = threadIdx.x;
    g[j] = pool[j] * inv_n;
    __syncthreads();
    float s = vb1[j];
    for (int c = 0; c < HID; ++c) s += g[c] * vW1[c * HID + j];
    s = fmaxf(s, 0.f);
    red[j] = s * vW2[j];
    __syncthreads();
    for (int off = 64; off; off >>= 1) {
        if (j < off) red[j] += red[j + off];
        __syncthreads();
    }
    if (j == 0) out[0] = red[0] + vb2[0];
}

// ---------------------------------------------------------------------------
// host orchestration
// ---------------------------------------------------------------------------
static void run_gat_layer(const float* Xin, int K, const float* W,
                          const float* a_s, const float* a_d, int heads,
                          const int* ei, float* Hf, float* Of, float* als,
                          float* ald, unsigned* mkey, float* den, float* alpha,
                          hipStream_t stream) {
    const int hshift = (heads == 4) ? 2 : 0;
    const int cshift = (heads == 4) ? 5 : 7;

    gemm_wmma_f32<<<Nn / 16, 256, 0, stream>>>(Xin, W, Hf, Nn, K);
    att_dots<<<(Nn + 7) / 8, 256, 0, stream>>>(Hf, a_s, a_d, als, ald, Nn, heads);

    fill_u<<<2048, 256, 0, stream>>>(mkey, 0u, (size_t)Nn * heads);
    fill_f<<<2048, 256, 0, stream>>>(den, 0.f, (size_t)Nn * heads);
    fill_f<<<8192, 256, 0, stream>>>(Of, 0.f, (size_t)Nn * HID);

    long long eh = (long long)ETOT << hshift;
    int blk12 = (int)((eh + 255) / 256);
    edge_logit_max<<<blk12, 256, 0, stream>>>(ei, als, ald, mkey, hshift);
    edge_exp_den<<<blk12, 256, 0, stream>>>(ei, als, ald, mkey, den, alpha, hshift);

    long long ec = (long long)ETOT << 7;
    int blk3 = (int)((ec + 255) / 256);
    edge_aggregate<<<blk3, 256, 0, stream>>>(ei, Hf, alpha, den, Of, cshift);
    // note: per-channel bias b_l is skipped on purpose — it is removed exactly
    // by the per-channel mean subtraction of the following BatchNorm.
}

static void run_bn_relu(float* X, const float* g, const float* b, float* accum,
                        float* sc, float* sh, hipStream_t stream) {
    fill_f<<<1, 256, 0, stream>>>(accum, 0.f, 2 * HID);
    col_stats<<<1024, HID, 0, stream>>>(X, accum, Nn, 1);
    bn_finalize<<<1, HID, 0, stream>>>(accum, g, b, sc, sh, 1.0f / Nn);
    bn_apply_relu<<<8192, 256, 0, stream>>>(X, sc, sh, Nn);
}

extern "C" void kernel_launch(void* const* d_in, const int* in_sizes, int n_in,
                              void* d_out, int out_size, void* d_ws, size_t ws_size,
                              hipStream_t stream) {
    (void)in_sizes; (void)n_in; (void)out_size; (void)ws_size;

    const float* x   = (const float*)d_in[0];
    const int*   ei  = (const int*)d_in[1];
    const float* W1  = (const float*)d_in[2];
    const float* as1 = (const float*)d_in[3];
    const float* ad1 = (const float*)d_in[4];
    const float* g1  = (const float*)d_in[6];
    const float* be1 = (const float*)d_in[7];
    const float* W2  = (const float*)d_in[8];
    const float* as2 = (const float*)d_in[9];
    const float* ad2 = (const float*)d_in[10];
    const float* g2  = (const float*)d_in[12];
    const float* be2 = (const float*)d_in[13];
    const float* W3  = (const float*)d_in[14];
    const float* as3 = (const float*)d_in[15];
    const float* ad3 = (const float*)d_in[16];
    const float* g3  = (const float*)d_in[18];
    const float* be3 = (const float*)d_in[19];
    const float* vW1 = (const float*)d_in[20];
    const float* vb1 = (const float*)d_in[21];
    const float* vW2 = (const float*)d_in[22];
    const float* vb2 = (const float*)d_in[23];
    float* out = (float*)d_out;

    // workspace carve-up (floats)
    float* ws = (float*)d_ws;
    float*    Hf    = ws;                                   // N*128
    float*    Of    = Hf + (size_t)Nn * HID;                // N*128
    float*    als   = Of + (size_t)Nn * HID;                // N*4
    float*    ald   = als + (size_t)Nn * 4;                 // N*4
    unsigned* mkey  = (unsigned*)(ald + (size_t)Nn * 4);    // N*4
    float*    den   = (float*)(mkey + (size_t)Nn * 4);      // N*4
    float*    alpha = den + (size_t)Nn * 4;                 // ETOT*4
    float*    accum = alpha + (size_t)ETOT * 4;             // 256
    float*    sc    = accum + 2 * HID;                      // 128
    float*    sh    = sc + HID;                             // 128
    float*    pool  = sh + HID;                             // 128

    // ---- layer 1: GAT(64 -> 4x32) + BN + ReLU (in/out: Of) ----
    run_gat_layer(x, FEAT, W1, as1, ad1, 4, ei, Hf, Of, als, ald, mkey, den,
                  alpha, stream);
    run_bn_relu(Of, g1, be1, accum, sc, sh, stream);

    // ---- layer 2: GAT(128 -> 4x32) + BN + ReLU ----
    // gemm reads Of into Hf; aggregation then overwrites Of (input dead after gemm)
    run_gat_layer(Of, HID, W2, as2, ad2, 4, ei, Hf, Of, als, ald, mkey, den,
                  alpha, stream);
    run_bn_relu(Of, g2, be2, accum, sc, sh, stream);

    // ---- layer 3: GAT(128 -> 1x128) + BN + ReLU ----
    run_gat_layer(Of, HID, W3, as3, ad3, 1, ei, Hf, Of, als, ald, mkey, den,
                  alpha, stream);
    run_bn_relu(Of, g3, be3, accum, sc, sh, stream);

    // ---- global mean pool + value head ----
    fill_f<<<1, 256, 0, stream>>>(pool, 0.f, HID);
    col_stats<<<1024, HID, 0, stream>>>(Of, pool, Nn, 0);
    value_head<<<1, HID, 0, stream>>>(pool, vW1, vb1, vW2, vb2, out, 1.0f / Nn);
}